// GATNet_45586782879798
// MI455X (gfx1250) — compile-verified
//
#include <hip/hip_runtime.h>
#include <math.h>

typedef float v2f __attribute__((ext_vector_type(2)));
typedef float v8f __attribute__((ext_vector_type(8)));

// -------- problem dims (fixed by reference setup_inputs) --------
constexpr int kB  = 4;       // batch
constexpr int kN0 = 20000;   // layer-1 src nodes
constexpr int kN1 = 15000;   // layer-1 dst / layer-2 src nodes
constexpr int kN2 = 10000;   // layer-2 dst nodes
constexpr int kH  = 4;       // heads
constexpr int kC1 = 16;      // layer-1 per-head channels
constexpr int kC2 = 32;      // layer-2 per-head channels
constexpr int kE0 = 320000;
constexpr int kE1 = 200000;
constexpr int kBH = kB * kH; // 16

// ---------------- fills ----------------
__global__ void fill_f32(float* p, int n, float v) {
    int t = blockIdx.x * blockDim.x + threadIdx.x;
    if (t < n) p[t] = v;
}
__global__ void fill_u32(unsigned* p, int n, unsigned v) {
    int t = blockIdx.x * blockDim.x + threadIdx.x;
    if (t < n) p[t] = v;
}

// ---------------- fp32 WMMA GEMM: C[M,N] = A[M,K] * B[K,N], all row-major --------------
// B (small weight matrix) staged in LDS once per block; each wave owns one 16-row
// A tile, preloads its full WMMA fragment set, and sweeps all N tiles so A is read
// exactly once. LDS row pad of +8 floats makes the two lane-halves (rows k / k+2)
// hit disjoint 16-bank groups of the 64-bank LDS (2*(N+8) % 64 == 16 for N=64,128).
template <int K, int N, int WAVES>
__global__ __launch_bounds__(32 * WAVES)
void gemm_wmma_f32(const float* __restrict__ A, const float* __restrict__ Bm,
                   float* __restrict__ C, int Mtiles)
{
    constexpr int NT   = N / 16;
    constexpr int PAD  = 8;
    constexpr int LDSN = N + PAD;
    __shared__ float sB[K * LDSN];

    const int tid = threadIdx.x;
    for (int i = tid; i < K * N; i += 32 * WAVES) {
        int k = i / N, n = i % N;
        sB[k * LDSN + n] = Bm[i];
    }
    __syncthreads();

    const int wave = tid >> 5;
    const int lane = tid & 31;      // wave32
    const int half = lane >> 4;     // 0: K pair {0,1}; 1: K pair {2,3}
    const int l16  = lane & 15;

    const int mtile = blockIdx.x * WAVES + wave;
    if (mtile >= Mtiles) return;    // wave-uniform: EXEC stays all-ones for live waves

    // preload full 16xK A tile in WMMA fragment layout: chunk c -> v2f at k=4c+2*half
    const float* arow = A + (size_t)(mtile * 16 + l16) * K + half * 2;
    v2f afrag[K / 4];
#pragma unroll
    for (int c = 0; c < K / 4; ++c) {
        afrag[c].x = arow[4 * c];
        afrag[c].y = arow[4 * c + 1];
    }

    float* crow = C + (size_t)(mtile * 16 + half * 8) * N + l16;
#pragma unroll
    for (int nt = 0; nt < NT; ++nt) {
        const int col = nt * 16 + l16;
        v8f acc = {};
#pragma unroll
        for (int c = 0; c < K / 4; ++c) {
            const int ka = 4 * c + half * 2;
            v2f b;
            b.x = sB[ka * LDSN + col];
            b.y = sB[(ka + 1) * LDSN + col];
            acc = __builtin_amdgcn_wmma_f32_16x16x4_f32(false, afrag[c], false, b,
                                                        (short)0, acc, false, false);
        }
#pragma unroll
        for (int v = 0; v < 8; ++v)
            crow[(size_t)v * N + nt * 16] = acc[v];
    }
}

// ---------------- per-node attention dots ----------------
// out[m*BH + b*H + h] = dot(XS[(b*NB + node)*H*Ch + h*Ch ..], att[h*2Ch + attOff ..])
__global__ void att_dots(const float* __restrict__ XS, const float* __restrict__ att,
                         const int* __restrict__ map, float* __restrict__ out,
                         int nOut, int NB, int Ch, int attOff)
{
    int t = blockIdx.x * blockDim.x + threadIdx.x;
    if (t >= nOut * kBH) return;
    int h = t % kH;
    int b = (t / kH) % kB;
    int m = t / kBH;
    int node = map ? map[m] : m;
    const float* row = XS + ((size_t)b * NB + node) * (size_t)(kH * Ch) + h * Ch;
    const float* a   = att + h * (2 * Ch) + attOff;
    float s = 0.f;
    for (int o = 0; o < Ch; ++o) s += row[o] * a[o];
    out[t] = s;
}

// ---------------- monotone float <-> u32 encoding for atomic max ----------------
__device__ __forceinline__ unsigned enc_f32(float x) {
    unsigned u = __float_as_uint(x);
    return (u & 0x80000000u) ? ~u : (u | 0x80000000u);
}
__device__ __forceinline__ float dec_f32(unsigned e) {
    unsigned u = (e & 0x80000000u) ? (e & 0x7FFFFFFFu) : ~e;
    return __uint_as_float(u);
}
constexpr unsigned kEncNegInf = 0x007FFFFFu;  // enc(-inf)

// ---------------- edge pass 1: segment max of leaky(alpha*ew) ----------------
__global__ void edge_max(const int* __restrict__ src, const int* __restrict__ dst,
                         const float* __restrict__ ew,
                         const float* __restrict__ aI, const float* __restrict__ aJ,
                         unsigned* __restrict__ amax, int E)
{
    int t = blockIdx.x * blockDim.x + threadIdx.x;
    if (t >= E * kBH) return;
    int bh = t % kBH;
    int e  = t / kBH;
    int d = dst[e], s = src[e];
    float al = (aI[(size_t)d * kBH + bh] + aJ[(size_t)s * kBH + bh]) * ew[e];
    al = (al > 0.f) ? al : al * 0.2f;               // leaky_relu, NEG_SLOPE=0.2
    atomicMax(&amax[(size_t)d * kBH + bh], enc_f32(al));
}

// ---------------- edge pass 2: accumulate denom + weighted numerator ----------------
__global__ void edge_acc(const int* __restrict__ src, const int* __restrict__ dst,
                         const float* __restrict__ ew,
                         const float* __restrict__ aI, const float* __restrict__ aJ,
                         const unsigned* __restrict__ amax,
                         const float* __restrict__ XS,
                         float* __restrict__ denom, float* __restrict__ num,
                         int E, int NB, int Ch)
{
    int t = blockIdx.x * blockDim.x + threadIdx.x;
    if (t >= E * kBH) return;
    int h = t % kH;
    int b = (t / kH) % kB;
    int e = t / kBH;
    int d = dst[e], s = src[e];
    int bh = b * kH + h;
    float al = (aI[(size_t)d * kBH + bh] + aJ[(size_t)s * kBH + bh]) * ew[e];
    al = (al > 0.f) ? al : al * 0.2f;
    float mx = dec_f32(amax[(size_t)d * kBH + bh]);
    if (!__builtin_isfinite(mx)) mx = 0.f;          // matches where(isfinite, amax, 0)
    float ex = expf(al - mx);
    atomicAdd(&denom[(size_t)d * kBH + bh], ex);
    const float* xj = XS  + ((size_t)b * NB + s) * (size_t)(kH * Ch) + h * Ch;
    float*       np = num + ((size_t)d * kB + b) * (size_t)(kH * Ch) + h * Ch;
    for (int o = 0; o < Ch; ++o)
        atomicAdd(&np[o], ex * xj[o]);
}

// ---------------- node finalize: mean over heads + bias + leaky(0.01) ----------------
// out[(b*Nd + m)*Ch + o]
__global__ void node_final(const float* __restrict__ num, const float* __restrict__ denom,
                           const float* __restrict__ bias, float* __restrict__ out,
                           int Nd, int Ch)
{
    int t = blockIdx.x * blockDim.x + threadIdx.x;
    if (t >= Nd * kB * Ch) return;
    int o = t % Ch;
    int b = (t / Ch) % kB;
    int m = t / (Ch * kB);
    float s = 0.f;
    for (int h = 0; h < kH; ++h) {
        float dn = denom[(size_t)m * kBH + b * kH + h] + 1e-16f;
        s += num[((size_t)m * kB + b) * (size_t)(kH * Ch) + h * Ch + o] / dn;
    }
    s = s * (1.0f / kH) + bias[o];
    s = (s > 0.f) ? s : s * 0.01f;
    out[((size_t)b * Nd + m) * Ch + o] = s;
}

// ---------------- host launcher ----------------
static inline int cdiv(long a, long b) { return (int)((a + b - 1) / b); }

extern "C" void kernel_launch(void* const* d_in, const int* in_sizes, int n_in,
                              void* d_out, int out_size, void* d_ws, size_t ws_size,
                              hipStream_t stream)
{
    (void)in_sizes; (void)n_in; (void)out_size; (void)ws_size;
    const float* X    = (const float*)d_in[0];
    const int*   src0 = (const int*)d_in[1];
    const int*   dst0 = (const int*)d_in[2];
    const float* ew0  = (const float*)d_in[3];
    const int*   res0 = (const int*)d_in[4];
    const int*   src1 = (const int*)d_in[5];
    const int*   dst1 = (const int*)d_in[6];
    const float* ew1  = (const float*)d_in[7];
    const int*   res1 = (const int*)d_in[8];
    const float* W1   = (const float*)d_in[9];
    const float* att1 = (const float*)d_in[10];
    const float* b1   = (const float*)d_in[11];
    const float* W2   = (const float*)d_in[12];
    const float* att2 = (const float*)d_in[13];
    const float* b2   = (const float*)d_in[14];

    float* ws = (float*)d_ws;
    size_t off = 0;
    float* XS1 = ws + off;  off += (size_t)kB * kN0 * kH * kC1;   // (B,N0,64) projected
    float* aJ1 = ws + off;  off += (size_t)kN0 * kBH;
    float* aI1 = ws + off;  off += (size_t)kN1 * kBH;
    unsigned* amax1 = (unsigned*)(ws + off); off += (size_t)kN1 * kBH;
    float* den1 = ws + off; off += (size_t)kN1 * kBH;
    float* num1 = ws + off; off += (size_t)kN1 * kB * kH * kC1;
    float* H1  = ws + off;  off += (size_t)kB * kN1 * kC1;        // (B,N1,16) layer-1 out
    float* XS2 = ws + off;  off += (size_t)kB * kN1 * kH * kC2;   // (B,N1,128) projected
    float* aJ2 = ws + off;  off += (size_t)kN1 * kBH;
    float* aI2 = ws + off;  off += (size_t)kN2 * kBH;
    unsigned* amax2 = (unsigned*)(ws + off); off += (size_t)kN2 * kBH;
    float* den2 = ws + off; off += (size_t)kN2 * kBH;
    float* num2 = ws + off; off += (size_t)kN2 * kB * kH * kC2;

    // ---- init accumulators (workspace is poisoned; re-init every call) ----
    fill_u32<<<cdiv((long)kN1 * kBH, 256), 256, 0, stream>>>(amax1, kN1 * kBH, kEncNegInf);
    fill_f32<<<cdiv((long)kN1 * kBH, 256), 256, 0, stream>>>(den1, kN1 * kBH, 0.f);
    fill_f32<<<cdiv((long)kN1 * kB * kH * kC1, 256), 256, 0, stream>>>(num1, kN1 * kB * kH * kC1, 0.f);
    fill_u32<<<cdiv((long)kN2 * kBH, 256), 256, 0, stream>>>(amax2, kN2 * kBH, kEncNegInf);
    fill_f32<<<cdiv((long)kN2 * kBH, 256), 256, 0, stream>>>(den2, kN2 * kBH, 0.f);
    fill_f32<<<cdiv((long)kN2 * kB * kH * kC2, 256), 256, 0, stream>>>(num2, kN2 * kB * kH * kC2, 0.f);

    // ---- layer 1 ----
    // X is (B,N0,64) row-major == A[M=B*N0, K=64]; W1 is (64,64)
    {
        const int mt = kB * kN0 / 16;                 // 5000 tiles
        gemm_wmma_f32<64, 64, 8><<<cdiv(mt, 8), 256, 0, stream>>>(X, W1, XS1, mt);
    }
    att_dots<<<cdiv((long)kN0 * kBH, 256), 256, 0, stream>>>(XS1, att1, nullptr, aJ1, kN0, kN0, kC1, kC1);
    att_dots<<<cdiv((long)kN1 * kBH, 256), 256, 0, stream>>>(XS1, att1, res0,    aI1, kN1, kN0, kC1, 0);
    edge_max<<<cdiv((long)kE0 * kBH, 256), 256, 0, stream>>>(src0, dst0, ew0, aI1, aJ1, amax1, kE0);
    edge_acc<<<cdiv((long)kE0 * kBH, 256), 256, 0, stream>>>(src0, dst0, ew0, aI1, aJ1, amax1,
                                                             XS1, den1, num1, kE0, kN0, kC1);
    node_final<<<cdiv((long)kN1 * kB * kC1, 256), 256, 0, stream>>>(num1, den1, b1, H1, kN1, kC1);

    // ---- layer 2 ----
    {
        const int mt = kB * kN1 / 16;                 // 3750 tiles
        gemm_wmma_f32<16, 128, 8><<<cdiv(mt, 8), 256, 0, stream>>>(H1, W2, XS2, mt);
    }
    att_dots<<<cdiv((long)kN1 * kBH, 256), 256, 0, stream>>>(XS2, att2, nullptr, aJ2, kN1, kN1, kC2, kC2);
    att_dots<<<cdiv((long)kN2 * kBH, 256), 256, 0, stream>>>(XS2, att2, res1,    aI2, kN2, kN1, kC2, 0);
    edge_max<<<cdiv((long)kE1 * kBH, 256), 256, 0, stream>>>(src1, dst1, ew1, aI2, aJ2, amax2, kE1);
    edge_acc<<<cdiv((long)kE1 * kBH, 256), 256, 0, stream>>>(src1, dst1, ew1, aI2, aJ2, amax2,
                                                             XS2, den2, num2, kE1, kN1, kC2);
    node_final<<<cdiv((long)kN2 * kB * kC2, 256), 256, 0, stream>>>(num2, den2, b2, (float*)d_out, kN2, kC2);
}